// AttnBlock_37366215475736
// MI455X (gfx1250) — compile-verified
//
#include <hip/hip_runtime.h>
#include <math.h>

typedef float v2f __attribute__((ext_vector_type(2)));
typedef float v8f __attribute__((ext_vector_type(8)));
typedef int   v4i __attribute__((ext_vector_type(4)));

static constexpr int kItems = 16;    // B*T
static constexpr int kC     = 512;
static constexpr int kHW    = 1024;  // H*W
static constexpr int kG     = 32;    // groups
static constexpr int kCpg   = 16;    // channels per group

// ---------------- CDNA5 async global->LDS copy (ASYNCcnt path) -------------
#if __has_builtin(__builtin_amdgcn_global_load_async_to_lds_b128) && \
    __has_builtin(__builtin_amdgcn_s_wait_asynccnt)
#define HAVE_ASYNC_LDS 1
typedef __attribute__((address_space(1))) v4i GlobalV4;
typedef __attribute__((address_space(3))) v4i LdsV4;
#else
#define HAVE_ASYNC_LDS 0
#endif

// copy 16 contiguous bytes global -> LDS, per lane (async, ASYNCcnt-tracked)
__device__ __forceinline__ void copy16_g2l(const float* __restrict__ g,
                                           float* l) {
#if HAVE_ASYNC_LDS
  __builtin_amdgcn_global_load_async_to_lds_b128((GlobalV4*)g, (LdsV4*)l, 0, 0);
#else
  *(float4*)l = *(const float4*)g;
#endif
}

__device__ __forceinline__ void wait_g2l() {
#if HAVE_ASYNC_LDS
  __builtin_amdgcn_s_wait_asynccnt(0);
#endif
}

// ---------------------------------------------------------------- GroupNorm
__global__ __launch_bounds__(256) void gn_kernel(const float* __restrict__ x,
                                                 const float* __restrict__ scale,
                                                 const float* __restrict__ bias,
                                                 float* __restrict__ h) {
  const int g    = blockIdx.x;
  const int item = blockIdx.y;
  const int tid  = threadIdx.x;
  const size_t base = ((size_t)item * kC + (size_t)g * kCpg) * kHW;
  const int n = kCpg * kHW;  // 16384

  float s = 0.f, ss = 0.f;
  for (int i = tid; i < n; i += 256) {
    float v = x[base + i];
    s += v;
    ss += v * v;
  }
  __shared__ float r0[256], r1[256];
  r0[tid] = s; r1[tid] = ss;
  __syncthreads();
  for (int st = 128; st > 0; st >>= 1) {
    if (tid < st) { r0[tid] += r0[tid + st]; r1[tid] += r1[tid + st]; }
    __syncthreads();
  }
  const float inv_n = 1.f / (float)n;
  const float mean  = r0[0] * inv_n;
  const float var   = r1[0] * inv_n - mean * mean;
  const float rstd  = rsqrtf(var + 1e-6f);
  for (int i = tid; i < n; i += 256) {
    const int ch = g * kCpg + (i >> 10);  // i / kHW
    const float v = x[base + i];
    h[base + i] = (v - mean) * rstd * scale[ch] + bias[ch];
  }
}

// ------------------------------------------------- generic strided WMMA GEMM
// D[m][n] = alpha * sum_k A[m][k]*B[k][n]  (+ bias[m]) (+ resid[m][n])
// Block tile 64(M) x 128(N), K chunk 16, 8 waves; wave (wm,wn) owns a 16x64
// strip as four 16x16 f32 WMMA accumulators. Double-buffered LDS tiles:
// async b128 copies of chunk i+1 overlap the 16 WMMAs of chunk i.
// AKF: A's k-stride == 1 in memory; BNF: B's n-stride == 1 in memory.
template <bool AKF, bool BNF>
__global__ __launch_bounds__(256) void gemm_wmma_f32(
    const float* __restrict__ A, const float* __restrict__ B,
    float* __restrict__ D,
    const float* __restrict__ bias, const float* __restrict__ resid,
    long long a_sm, long long a_sk, long long a_sb,
    long long b_sk, long long b_sn, long long b_sb,
    long long d_sm, long long d_sb,
    float alpha, int K) {
  // LDS layouts chosen so global b128 async copies land contiguously and
  // fragment reads dodge bank conflicts between the two lane halves.
  __shared__ float As[2][AKF ? 64 * 20 : 16 * 72];
  __shared__ float Bs[2][BNF ? 16 * 136 : 128 * 20];

  const int tid  = threadIdx.x;
  const int lane = tid & 31;
  const int wave = tid >> 5;   // 0..7
  const int wm   = wave >> 1;  // 0..3  (M strip)
  const int wn   = wave & 1;   // 0..1  (N half)
  const int hi   = lane >> 4;  // K-half selector within fragment
  const int l16  = lane & 15;

  const int item = blockIdx.z;
  const int M0 = blockIdx.y * 64;
  const int N0 = blockIdx.x * 128;

  A += (size_t)item * a_sb;
  B += (size_t)item * b_sb;
  D += (size_t)item * d_sb;
  if (resid) resid += (size_t)item * d_sb;

  const int ml  = wm * 16 + l16;  // row in A tile
  const int nl0 = wn * 64 + l16;  // base col in B tile

  // issue this thread's 3 async b128 copies for K-chunk starting at k0
  auto issue_tiles = [&](float* AsBuf, float* BsBuf, int k0) {
    // ---- A tile (64m x 16k): one b128 per thread
    if (AKF) {
      const int m = tid & 63, k4 = (tid >> 6) << 2;
      copy16_g2l(A + (size_t)(M0 + m) * a_sm + (size_t)(k0 + k4) * a_sk,
                 &AsBuf[m * 20 + k4]);
    } else {  // m-fast in memory (a_sm == 1)
      const int k = tid & 15, m4 = (tid >> 4) << 2;
      copy16_g2l(A + (size_t)(M0 + m4) * a_sm + (size_t)(k0 + k) * a_sk,
                 &AsBuf[k * 72 + m4]);
    }
    // ---- B tile (16k x 128n): two b128 per thread
#pragma unroll
    for (int it = 0; it < 2; ++it) {
      const int li = tid + it * 256;
      if (BNF) {
        const int k = li >> 5, n4 = (li & 31) << 2;
        copy16_g2l(B + (size_t)(k0 + k) * b_sk + (size_t)(N0 + n4) * b_sn,
                   &BsBuf[k * 136 + n4]);
      } else {  // k-fast in memory (b_sk == 1)
        const int n = li & 127, k4 = (li >> 7) << 2;
        copy16_g2l(B + (size_t)(k0 + k4) * b_sk + (size_t)(N0 + n) * b_sn,
                   &BsBuf[n * 20 + k4]);
      }
    }
  };

  v8f acc[4];
#pragma unroll
  for (int t = 0; t < 4; ++t)
#pragma unroll
    for (int i = 0; i < 8; ++i) acc[t][i] = 0.f;

  issue_tiles(As[0], Bs[0], 0);  // prologue: fill buffer 0
  int cur = 0;

  for (int k0 = 0; k0 < K; k0 += 16) {
    wait_g2l();       // my copies into buf[cur] have landed
    __syncthreads();  // everyone's copies landed; buf[cur^1] no longer read
    if (k0 + 16 < K) issue_tiles(As[cur ^ 1], Bs[cur ^ 1], k0 + 16);

    const float* Asb = As[cur];
    const float* Bsb = Bs[cur];
#pragma unroll
    for (int kk = 0; kk < 16; kk += 4) {
      const int kb = kk + 2 * hi;
      // A fragment 16x4 f32: lanes 0-15 K={kk,kk+1}, lanes 16-31 K={kk+2,kk+3}
      v2f a;
      if (AKF) {
        a.x = Asb[ml * 20 + kb];
        a.y = Asb[ml * 20 + kb + 1];
      } else {
        a.x = Asb[kb * 72 + ml];
        a.y = Asb[(kb + 1) * 72 + ml];
      }
#pragma unroll
      for (int t = 0; t < 4; ++t) {
        const int n = nl0 + 16 * t;
        v2f bf;
        if (BNF) {
          bf.x = Bsb[kb * 136 + n];
          bf.y = Bsb[(kb + 1) * 136 + n];
        } else {
          bf.x = Bsb[n * 20 + kb];
          bf.y = Bsb[n * 20 + kb + 1];
        }
        acc[t] = __builtin_amdgcn_wmma_f32_16x16x4_f32(
            false, a, false, bf, (short)0, acc[t], false, false);
      }
    }
    cur ^= 1;
  }

  // epilogue: D layout — VGPR i holds rows (i, i+8) across lane halves
#pragma unroll
  for (int t = 0; t < 4; ++t) {
#pragma unroll
    for (int i = 0; i < 8; ++i) {
      const int m = M0 + wm * 16 + i + 8 * hi;
      const int n = N0 + wn * 64 + 16 * t + l16;
      const size_t off = (size_t)m * d_sm + n;
      float v = acc[t][i] * alpha;
      if (bias)  v += bias[m];
      if (resid) v += resid[off];
      D[off] = v;
    }
  }
}

// ------------------------------------------------------------- row softmax
__global__ __launch_bounds__(256) void softmax_kernel(float* __restrict__ w) {
  const int row  = blockIdx.x;
  const int item = blockIdx.y;
  const int tid  = threadIdx.x;
  float* p = w + ((size_t)item * kHW + row) * kHW;

  float r[4];
  float m = -INFINITY;
#pragma unroll
  for (int j = 0; j < 4; ++j) {
    r[j] = p[tid + 256 * j];
    m = fmaxf(m, r[j]);
  }
  __shared__ float red[256];
  red[tid] = m;
  __syncthreads();
  for (int st = 128; st > 0; st >>= 1) {
    if (tid < st) red[tid] = fmaxf(red[tid], red[tid + st]);
    __syncthreads();
  }
  m = red[0];
  __syncthreads();

  float s = 0.f;
#pragma unroll
  for (int j = 0; j < 4; ++j) {
    r[j] = __expf(r[j] - m);
    s += r[j];
  }
  red[tid] = s;
  __syncthreads();
  for (int st = 128; st > 0; st >>= 1) {
    if (tid < st) red[tid] += red[tid + st];
    __syncthreads();
  }
  const float inv = 1.f / red[0];
#pragma unroll
  for (int j = 0; j < 4; ++j) p[tid + 256 * j] = r[j] * inv;
}

// ------------------------------------------------------------------ launch
extern "C" void kernel_launch(void* const* d_in, const int* in_sizes, int n_in,
                              void* d_out, int out_size, void* d_ws, size_t ws_size,
                              hipStream_t stream) {
  (void)in_sizes; (void)n_in; (void)out_size; (void)ws_size;
  const float* x   = (const float*)d_in[0];
  const float* gns = (const float*)d_in[1];
  const float* gnb = (const float*)d_in[2];
  const float* wq  = (const float*)d_in[3];
  const float* bq  = (const float*)d_in[4];
  const float* wk  = (const float*)d_in[5];
  const float* bk  = (const float*)d_in[6];
  const float* wv  = (const float*)d_in[7];
  const float* bv  = (const float*)d_in[8];
  const float* wp  = (const float*)d_in[9];
  const float* bp  = (const float*)d_in[10];
  float* out = (float*)d_out;

  const long long PLANE  = (long long)kC * kHW;   // 524288
  const long long SPLANE = (long long)kHW * kHW;  // 1048576

  float* h  = (float*)d_ws;                       // 32 MB (reused for attn_out)
  float* q  = h  + (size_t)kItems * PLANE;        // 32 MB
  float* kb = q  + (size_t)kItems * PLANE;        // 32 MB
  float* v  = kb + (size_t)kItems * PLANE;        // 32 MB
  float* sc = v  + (size_t)kItems * PLANE;        // 64 MB

  const float attn_scale = 0.04419417382415922f;  // 512^-0.5

  // 1. GroupNorm
  gn_kernel<<<dim3(kG, kItems), 256, 0, stream>>>(x, gns, gnb, h);

  // 2. q/k/v = W(512x512) * h(512x1024) + b   (A row-major, B row-major)
  gemm_wmma_f32<true, true><<<dim3(8, 8, kItems), 256, 0, stream>>>(
      wq, h, q, bq, nullptr, 512, 1, 0, 1024, 1, PLANE, 1024, PLANE, 1.f, 512);
  gemm_wmma_f32<true, true><<<dim3(8, 8, kItems), 256, 0, stream>>>(
      wk, h, kb, bk, nullptr, 512, 1, 0, 1024, 1, PLANE, 1024, PLANE, 1.f, 512);
  gemm_wmma_f32<true, true><<<dim3(8, 8, kItems), 256, 0, stream>>>(
      wv, h, v, bv, nullptr, 512, 1, 0, 1024, 1, PLANE, 1024, PLANE, 1.f, 512);

  // 3. scores[i][j] = alpha * sum_c q[c][i]*k[c][j]   (A = q^T via strides)
  gemm_wmma_f32<false, true><<<dim3(8, 16, kItems), 256, 0, stream>>>(
      q, kb, sc, nullptr, nullptr,
      1, 1024, PLANE, 1024, 1, PLANE, 1024, SPLANE, attn_scale, 512);

  // 4. softmax over j
  softmax_kernel<<<dim3(kHW, kItems), 256, 0, stream>>>(sc);

  // 5. attn_out[c][i] = sum_j v[c][j]*w[i][j]   (B = w^T via strides) -> h
  gemm_wmma_f32<true, false><<<dim3(8, 8, kItems), 256, 0, stream>>>(
      v, sc, h, nullptr, nullptr,
      1024, 1, PLANE, 1, 1024, SPLANE, 1024, PLANE, 1.f, 1024);

  // 6. out = x + Wproj * attn_out + bproj
  gemm_wmma_f32<true, true><<<dim3(8, 8, kItems), 256, 0, stream>>>(
      wp, h, out, bp, x, 512, 1, 0, 1024, 1, PLANE, 1024, PLANE, 1.f, 512);
}